// GNNEncoder_12266426598044
// MI455X (gfx1250) — compile-verified
//
#include <hip/hip_runtime.h>

#define N_NODES 50000
#define N_EDGES 800000
#define FD 128   // feature dim

typedef float v2f __attribute__((ext_vector_type(2)));
typedef float v8f __attribute__((ext_vector_type(8)));

// ---------------------------------------------------------------------------
// Zero a float buffer (grid-stride).
// ---------------------------------------------------------------------------
__global__ void zero_f32_kernel(float* __restrict__ p, int n) {
    int i = blockIdx.x * blockDim.x + threadIdx.x;
    int stride = gridDim.x * blockDim.x;
    for (; i < n; i += stride) p[i] = 0.0f;
}

// ---------------------------------------------------------------------------
// Degree: deg[dst[e]] += 1.0  (float so the max(deg,1) divide matches ref)
// ---------------------------------------------------------------------------
__global__ void degree_kernel(const int* __restrict__ dst, float* __restrict__ deg) {
    int i = blockIdx.x * blockDim.x + threadIdx.x;
    int stride = gridDim.x * blockDim.x;
    for (; i < N_EDGES; i += stride) {
        atomicAdd(&deg[dst[i]], 1.0f);
    }
}

// ---------------------------------------------------------------------------
// Edge scatter: one wave32 per edge. Each lane moves a float4 (32*16B = 512B
// = one full 128-float feature row), then 4 global_atomic_add_f32 into agg.
// Working set (h + agg ~= 51MB) is resident in the 192MB L2, so these
// atomics resolve at L2 rate, not HBM.
// ---------------------------------------------------------------------------
__global__ void __launch_bounds__(256)
scatter_kernel(const float* __restrict__ h,
               const int* __restrict__ src,
               const int* __restrict__ dst,
               float* __restrict__ agg) {
    int gtid = blockIdx.x * blockDim.x + threadIdx.x;
    int edge = gtid >> 5;           // one wave32 per edge
    int lane = gtid & 31;
    if (edge >= N_EDGES) return;

    int s = src[edge];
    int d = dst[edge];

    const float4 v = *((const float4*)(h + (size_t)s * FD) + lane);
    float* ap = agg + (size_t)d * FD + lane * 4;
    atomicAdd(ap + 0, v.x);
    atomicAdd(ap + 1, v.y);
    atomicAdd(ap + 2, v.z);
    atomicAdd(ap + 3, v.w);
}

// ---------------------------------------------------------------------------
// Fused SAGE layer GEMM:
//   out = relu( (agg/max(deg,1)) @ Wl^T + b + h @ Wr^T )
//
// Block = 256 threads = 8 waves. blockIdx.x picks a 16-row node tile
// (50000 = 3125*16 exactly); wave id picks one of the 8 16-col output tiles.
// Each wave runs 2 x 32 v_wmma_f32_16x16x4_f32 steps (full fp32 precision).
// Degree normalization is folded into the A-operand load (GEMM is linear).
//
// Operand layouts (ISA 7.12.2, wave32):
//   A 16x4 f32 : lane m=lane&15, half=lane>>4; VGPR0=K(2*half), VGPR1=K(2*half+1)
//   B 4x16 f32 : mirrored, with B[k][n] = W[n][k] (row n = lane&15 of W)
//   C/D 16x16  : VGPR r -> row r + 8*half, col lane&15
// ---------------------------------------------------------------------------
__global__ void __launch_bounds__(256)
sage_gemm_relu_kernel(const float* __restrict__ agg,   // raw neighbor sums [N,FD]
                      const float* __restrict__ deg,   // [N]
                      const float* __restrict__ h,     // self features [N,FD]
                      const float* __restrict__ Wl,    // [FD,FD] row-major [out][in]
                      const float* __restrict__ Wr,    // [FD,FD]
                      const float* __restrict__ bias,  // [FD]
                      float* __restrict__ out) {       // [N,FD]
    const int wave   = threadIdx.x >> 5;    // 0..7 : output-column tile
    const int lane   = threadIdx.x & 31;
    const int half   = lane >> 4;           // 0/1 : K half for A/B operands
    const int l16    = lane & 15;
    const int m_base = blockIdx.x * 16;
    const int n_base = wave * 16;

    const int m = m_base + l16;             // A row this lane feeds
    const int n = n_base + l16;             // W row this lane feeds (B = W^T)

    const float invdeg = 1.0f / fmaxf(deg[m], 1.0f);

    const float* __restrict__ arow  = agg + (size_t)m * FD + 2 * half;
    const float* __restrict__ hrow  = h   + (size_t)m * FD + 2 * half;
    const float* __restrict__ blrow = Wl  + (size_t)n * FD + 2 * half;
    const float* __restrict__ brrow = Wr  + (size_t)n * FD + 2 * half;

    v8f acc = {};

    // --- agg (normalized) @ Wl^T ---
#pragma unroll
    for (int k = 0; k < FD; k += 4) {
        float2 av = *(const float2*)(arow + k);
        float2 bv = *(const float2*)(blrow + k);
        v2f a; a.x = av.x * invdeg; a.y = av.y * invdeg;
        v2f b; b.x = bv.x;          b.y = bv.y;
        acc = __builtin_amdgcn_wmma_f32_16x16x4_f32(
                  false, a, false, b, (short)0, acc, false, false);
    }

    // --- h @ Wr^T (accumulate into same C) ---
#pragma unroll
    for (int k = 0; k < FD; k += 4) {
        float2 av = *(const float2*)(hrow + k);
        float2 bv = *(const float2*)(brrow + k);
        v2f a; a.x = av.x; a.y = av.y;
        v2f b; b.x = bv.x; b.y = bv.y;
        acc = __builtin_amdgcn_wmma_f32_16x16x4_f32(
                  false, a, false, b, (short)0, acc, false, false);
    }

    // Epilogue: bias (per output column) + ReLU, store per C/D layout.
    const float bcol = bias[n_base + l16];
#pragma unroll
    for (int r = 0; r < 8; ++r) {
        const int row = m_base + r + 8 * half;
        const float v = fmaxf(acc[r] + bcol, 0.0f);
        out[(size_t)row * FD + n_base + l16] = v;
    }
}

// ---------------------------------------------------------------------------
// Host-side orchestration
// ---------------------------------------------------------------------------
static void run_layer(const float* h_in, const int* src, const int* dst,
                      const float* Wl, const float* Wr, const float* b,
                      float* deg, float* agg, float* h_out, hipStream_t stream) {
    const int aggN = N_NODES * FD;
    zero_f32_kernel<<<2048, 256, 0, stream>>>(agg, aggN);
    {
        // one wave32 per edge -> E*32 threads
        const long long threads = (long long)N_EDGES * 32;
        const int blocks = (int)((threads + 255) / 256);
        scatter_kernel<<<blocks, 256, 0, stream>>>(h_in, src, dst, agg);
    }
    sage_gemm_relu_kernel<<<N_NODES / 16, 256, 0, stream>>>(
        agg, deg, h_in, Wl, Wr, b, h_out);
}

extern "C" void kernel_launch(void* const* d_in, const int* in_sizes, int n_in,
                              void* d_out, int out_size, void* d_ws, size_t ws_size,
                              hipStream_t stream) {
    // Inputs (setup_inputs order):
    // 0: x [N,FD] f32, 1: edge_index [2,E] i32,
    // 2..10: W_l1,W_r1,b1, W_l2,W_r2,b2, W_l3,W_r3,b3
    const float* x   = (const float*)d_in[0];
    const int* ei    = (const int*)d_in[1];
    const int* src   = ei;            // edge_index[0]
    const int* dst   = ei + N_EDGES;  // edge_index[1]
    const float* Wl1 = (const float*)d_in[2];
    const float* Wr1 = (const float*)d_in[3];
    const float* b1  = (const float*)d_in[4];
    const float* Wl2 = (const float*)d_in[5];
    const float* Wr2 = (const float*)d_in[6];
    const float* b2  = (const float*)d_in[7];
    const float* Wl3 = (const float*)d_in[8];
    const float* Wr3 = (const float*)d_in[9];
    const float* b3  = (const float*)d_in[10];
    float* out = (float*)d_out;

    // Workspace layout: deg[N] | agg[N*FD] | h1[N*FD] | h2[N*FD]  (~77 MB)
    float* deg = (float*)d_ws;
    float* agg = deg + N_NODES;
    float* h1  = agg + (size_t)N_NODES * FD;
    float* h2  = h1  + (size_t)N_NODES * FD;

    // Degree (recomputed every call: deterministic, no cached state).
    zero_f32_kernel<<<256, 256, 0, stream>>>(deg, N_NODES);
    degree_kernel<<<3125, 256, 0, stream>>>(dst, deg);

    run_layer(x,  src, dst, Wl1, Wr1, b1, deg, agg, h1,  stream);
    run_layer(h1, src, dst, Wl2, Wr2, b2, deg, agg, h2,  stream);
    run_layer(h2, src, dst, Wl3, Wr3, b3, deg, agg, out, stream);
}